// EmbeddingOnehot_26800595927614
// MI455X (gfx1250) — compile-verified
//
#include <hip/hip_runtime.h>
#include <hip/hip_bf16.h>

// Embedding via one-hot matmul == row gather: out[t,:] = W[idx[t],:].
// Memory-bound: ~16 MB total traffic -> ~0.7us at 23.3 TB/s HBM. Zero FLOPs;
// WMMA would inflate this to a 134 GFLOP dense matmul plus a 2 GB one-hot, so
// the optimal CDNA5 kernel uses the memory paths instead:
//   - per-wave scalarized token index (v_readfirstlane -> s_load_b32)
//   - saddr-form global_load_b128 / global_store_b128, fully coalesced
//     (one wave32 = one 512B row: 32 lanes x float4)
//   - 4x unroll for memory-level parallelism (clauses of b128 loads)
//   - global_prefetch_b8 (WGP scope) on the streaming index array

#ifndef UNROLL
#define UNROLL 4
#endif

// Fast path: VECTOR_SIZE == 128 floats == 32 float4 == exactly one wave32 row.
// gid is the flat float4 index of the output; token = gid >> 5 is uniform
// across each wave; (gid>>5)*32 + (gid&31) == gid so stores are contiguous.
__global__ __launch_bounds__(256) void embed_gather128_kernel(
    const int* __restrict__ idx,
    const float4* __restrict__ w,    // weights viewed as rows of 32 float4
    float4* __restrict__ out,        // output viewed as rows of 32 float4
    unsigned n_tokens)
{
    const unsigned lane = threadIdx.x & 31u;
    const unsigned gid0 = blockIdx.x * (256u * UNROLL) + threadIdx.x;

    // Prefetch the index stream ahead, with a wave-uniform (all-SALU) clamp.
    {
        unsigned pf_tok =
            (unsigned)__builtin_amdgcn_readfirstlane((int)(gid0 >> 5)) + 2048u;
        if (pf_tok >= n_tokens) pf_tok = n_tokens - 1u;
        __builtin_prefetch(&idx[pf_tok], 0, 3);   // WGP-scope global_prefetch_b8
    }

    float4   v[UNROLL];
    unsigned g[UNROLL];

#pragma unroll
    for (int u = 0; u < UNROLL; ++u) {
        const unsigned gid = gid0 + (unsigned)u * 256u;
        g[u] = gid;
        // Token id is wave-uniform: force into SGPR so the index fetch is an
        // s_load_b32 and the row base is scalar (saddr-form global_load_b128).
        const unsigned token = (unsigned)__builtin_amdgcn_readfirstlane((int)(gid >> 5));
        const int row = idx[token];
        v[u] = w[(size_t)row * 32u + lane];       // 16B/lane = 512B/wave row
    }

#pragma unroll
    for (int u = 0; u < UNROLL; ++u) {
        out[g[u]] = v[u];                         // contiguous global_store_b128
    }
}

// Generic fallback for any (n, vec) shape: one block per token, no div/mod.
__global__ void embed_gather_generic_kernel(
    const int* __restrict__ idx,
    const float* __restrict__ w,
    float* __restrict__ out,
    long vec)
{
    const long t   = (long)blockIdx.x;
    const long row = (long)idx[t];
    for (long c = threadIdx.x; c < vec; c += blockDim.x) {
        out[t * vec + c] = w[row * vec + c];
    }
}

extern "C" void kernel_launch(void* const* d_in, const int* in_sizes, int n_in,
                              void* d_out, int out_size, void* d_ws, size_t ws_size,
                              hipStream_t stream) {
    (void)n_in; (void)d_ws; (void)ws_size;

    const int*   idx = (const int*)d_in[0];    // [n_tokens] int32
    const float* w   = (const float*)d_in[1];  // [word_size, vec] float32
    float*       out = (float*)d_out;          // [n_tokens, vec] float32

    const long n_tokens = (long)in_sizes[0];
    const long vec      = (n_tokens > 0) ? ((long)out_size / n_tokens) : 0;

    if (vec == 128 && n_tokens > 0) {
        const unsigned total_vec4 = (unsigned)(n_tokens * 32);
        const unsigned per_block  = 256u * UNROLL;
        if (total_vec4 % per_block == 0) {       // 16384 tokens -> 512 blocks
            const unsigned blocks = total_vec4 / per_block;
            embed_gather128_kernel<<<blocks, 256, 0, stream>>>(
                idx, (const float4*)w, (float4*)out, (unsigned)n_tokens);
            return;
        }
    }

    // Fallback: one block per token, threads stride the embedding vector.
    if (n_tokens > 0 && vec > 0) {
        embed_gather_generic_kernel<<<(unsigned)n_tokens, 256, 0, stream>>>(
            idx, w, out, vec);
    }
}